// Input_Embed_21534966022871
// MI455X (gfx1250) — compile-verified
//
#include <hip/hip_runtime.h>

#define B_ 32
#define N_ 2048
#define K_ 20
#define E_ 64

constexpr int NK  = N_ * K_;
constexpr int TOT = B_ * N_ * K_;   // 1,310,720 edge vectors

typedef __attribute__((ext_vector_type(2))) float v2f;
typedef __attribute__((ext_vector_type(8))) float v8f;

// ---------------------------------------------------------------------------
// Kernel 1: fused KNN (no materialized distance matrix).
// One thread = one query row; candidates staged through LDS in 256-pt tiles;
// register-resident top-20 via unrolled insertion (all static indices).
// pdist = -(||xi-xj||^2) so top-K of pdist == K nearest (self included, as ref).
// ---------------------------------------------------------------------------
__global__ __launch_bounds__(256) void knn_kernel(const float* __restrict__ xyz,
                                                  int* __restrict__ idxws) {
    __shared__ float sx[256], sy[256], sz[256];
    const int tid = threadIdx.x;
    const int b   = blockIdx.y;
    const int n   = blockIdx.x * 256 + tid;
    const float* pb = xyz + (size_t)b * N_ * 3;
    const float cx = pb[n * 3 + 0];
    const float cy = pb[n * 3 + 1];
    const float cz = pb[n * 3 + 2];

    float bd[K_];
    int   bi[K_];
#pragma unroll
    for (int j = 0; j < K_; ++j) { bd[j] = -3.402823466e38f; bi[j] = 0; }

    for (int tile = 0; tile < N_; tile += 256) {
        __syncthreads();
        sx[tid] = pb[(tile + tid) * 3 + 0];
        sy[tid] = pb[(tile + tid) * 3 + 1];
        sz[tid] = pb[(tile + tid) * 3 + 2];
        __syncthreads();
#pragma unroll 4
        for (int t = 0; t < 256; ++t) {
            const float dx = cx - sx[t];
            const float dy = cy - sy[t];
            const float dz = cz - sz[t];
            const float pd = -(dx * dx + dy * dy + dz * dz);
            if (pd > bd[K_ - 1]) {
                bd[K_ - 1] = pd; bi[K_ - 1] = tile + t;
#pragma unroll
                for (int j = K_ - 1; j > 0; --j) {
                    if (bd[j] > bd[j - 1]) {
                        float tf = bd[j]; bd[j] = bd[j - 1]; bd[j - 1] = tf;
                        int   ti = bi[j]; bi[j] = bi[j - 1]; bi[j - 1] = ti;
                    }
                }
            }
        }
    }
    int* o = idxws + ((size_t)b * N_ + n) * K_;
#pragma unroll
    for (int j = 0; j < K_; ++j) o[j] = bi[j];
}

// ---------------------------------------------------------------------------
// Kernel 2: edge-feature first/second moments (6 + 21 values).
// BN mean/var per channel are derived algebraically from these in kernel 3,
// so the 335 MB intermediate h is never materialized or re-read.
// ---------------------------------------------------------------------------
__global__ __launch_bounds__(256) void stats_kernel(const float* __restrict__ xyz,
                                                    const int* __restrict__ idxws,
                                                    float* __restrict__ stats) {
    const int gtid   = blockIdx.x * blockDim.x + threadIdx.x;
    const int stride = gridDim.x * blockDim.x;
    float acc[27];
#pragma unroll
    for (int t = 0; t < 27; ++t) acc[t] = 0.f;

    for (int i = gtid; i < TOT; i += stride) {
        const int b   = i / NK;
        const int rem = i - b * NK;
        const int n   = rem / K_;
        const int nb  = idxws[i];
        const float* pc = xyz + ((size_t)b * N_ + n)  * 3;
        const float* pn = xyz + ((size_t)b * N_ + nb) * 3;
        float e[6];
        e[0] = pc[0]; e[1] = pc[1]; e[2] = pc[2];
        e[3] = pn[0] - e[0]; e[4] = pn[1] - e[1]; e[5] = pn[2] - e[2];
#pragma unroll
        for (int c = 0; c < 6; ++c) acc[c] += e[c];
        int t = 6;
#pragma unroll
        for (int c = 0; c < 6; ++c)
#pragma unroll
            for (int c2 = c; c2 < 6; ++c2) { acc[t] += e[c] * e[c2]; ++t; }
    }
#pragma unroll
    for (int t = 0; t < 27; ++t) {
        float v = acc[t];
        for (int off = 16; off; off >>= 1) v += __shfl_xor(v, off, 32);
        if ((threadIdx.x & 31) == 0) atomicAdd(&stats[t], v);
    }
}

// ---------------------------------------------------------------------------
// Kernel 3: per-channel BN affine (scale/shift) from 6-vec mean + 6x6 moment.
// ---------------------------------------------------------------------------
__global__ __launch_bounds__(64) void finalize_kernel(const float* __restrict__ W,
                                                      const float* __restrict__ gamma,
                                                      const float* __restrict__ beta,
                                                      const float* __restrict__ stats,
                                                      float* __restrict__ scale,
                                                      float* __restrict__ shift) {
    const int e = threadIdx.x;
    const float inv = 1.0f / (float)TOT;
    float m[6];
#pragma unroll
    for (int c = 0; c < 6; ++c) m[c] = stats[c] * inv;
    float S[6][6];
    int t = 6;
#pragma unroll
    for (int c = 0; c < 6; ++c)
#pragma unroll
        for (int c2 = c; c2 < 6; ++c2) {
            const float s = stats[t] * inv;
            S[c][c2] = s; S[c2][c] = s; ++t;
        }
    float w[6];
#pragma unroll
    for (int c = 0; c < 6; ++c) w[c] = W[e * 6 + c];
    float me = 0.f, ex2 = 0.f;
#pragma unroll
    for (int c = 0; c < 6; ++c) {
        me += w[c] * m[c];
#pragma unroll
        for (int c2 = 0; c2 < 6; ++c2) ex2 += w[c] * w[c2] * S[c][c2];
    }
    const float var = ex2 - me * me;
    const float sc  = gamma[e] * rsqrtf(var + 1e-5f);
    scale[e] = sc;
    shift[e] = beta[e] - me * sc;
}

__device__ __forceinline__ float hsw(float x) {
    const float c = fminf(fmaxf(x + 3.0f, 0.0f), 6.0f);
    return x * c * (1.0f / 6.0f);
}

// ---------------------------------------------------------------------------
// Kernel 4: one wave per point. edge[K=20 (pad 32), 6 (pad 8)] x W^T[8,64]
// via V_WMMA_F32_16X16X4_F32: 2 M-tiles x 4 N-tiles x 2 K-steps = 16 WMMAs.
// A 16x4 f32 fragment layout (ISA 7.12.2): VGPR0 = K{0|2}, VGPR1 = K{1|3}
// across lane halves; B mirrors with N=lane%16. D: VGPRj -> rows j / j+8.
// BN affine + hardswish applied per element, then max over K folds into an
// 8-wide per-lane max + shfl_xor(16). Rows 20..31 of M-tile 1 never read.
// ---------------------------------------------------------------------------
__global__ __launch_bounds__(32) void edgeconv_kernel(const float* __restrict__ xyz,
                                                      const float* __restrict__ W,
                                                      const int* __restrict__ idxws,
                                                      const float* __restrict__ scale,
                                                      const float* __restrict__ shift,
                                                      float* __restrict__ out) {
    const int lane = threadIdx.x;
    const int r    = lane & 15;
    const int h    = lane >> 4;
    const int gid  = blockIdx.x;
    const int b    = gid >> 11;          // N_ == 2048
    const int n    = gid & (N_ - 1);

    const int pbase = (b * N_ + n) * 3;
    const float cx = xyz[pbase + 0];
    const float cy = xyz[pbase + 1];
    const float cz = xyz[pbase + 2];
    if (lane < 3) out[pbase + lane] = xyz[pbase + lane];   // xyz passthrough

    const int ibase = (b * N_ + n) * K_;
    // M-tile 0: k = r (0..15, always valid). M-tile 1: k = 16+r (valid r<4).
    const int  i0 = idxws[ibase + r];
    const bool v1 = (r < 4);
    const int  i1 = v1 ? idxws[ibase + 16 + r] : i0;
    const float* p0 = xyz + (b * N_ + i0) * 3;
    const float* p1 = xyz + (b * N_ + i1) * 3;
    const float d0x = p0[0] - cx, d0y = p0[1] - cy, d0z = p0[2] - cz;
    const float e1c0 = v1 ? cx : 0.f, e1c1 = v1 ? cy : 0.f, e1c2 = v1 ? cz : 0.f;
    const float d1x = v1 ? (p1[0] - cx) : 0.f;
    const float d1y = v1 ? (p1[1] - cy) : 0.f;
    const float d1z = v1 ? (p1[2] - cz) : 0.f;

    // A fragments: edge row features c = {2h,2h+1} (K-step0), {4+2h,5+2h} (K-step1; h==1 -> pad 0)
    const v2f A00 = { h ? cz   : cx,   h ? d0x : cy   };
    const v2f A01 = { h ? 0.f  : d0y,  h ? 0.f : d0z  };
    const v2f A10 = { h ? e1c2 : e1c0, h ? d1x : e1c1 };
    const v2f A11 = { h ? 0.f  : d1y,  h ? 0.f : d1z  };

    float* outx = out + (size_t)B_ * N_ * 3;

#pragma unroll
    for (int nt = 0; nt < 4; ++nt) {
        const int ech = nt * 16 + r;
        const int wb  = ech * 6;
        const v2f B0 = { W[wb + 2 * h], W[wb + 2 * h + 1] };
        const v2f B1 = { h ? 0.f : W[wb + 4], h ? 0.f : W[wb + 5] };
        v8f acc0 = {0.f, 0.f, 0.f, 0.f, 0.f, 0.f, 0.f, 0.f};
        v8f acc1 = {0.f, 0.f, 0.f, 0.f, 0.f, 0.f, 0.f, 0.f};
        acc0 = __builtin_amdgcn_wmma_f32_16x16x4_f32(false, A00, false, B0, (short)0, acc0, false, false);
        acc0 = __builtin_amdgcn_wmma_f32_16x16x4_f32(false, A01, false, B1, (short)0, acc0, false, false);
        acc1 = __builtin_amdgcn_wmma_f32_16x16x4_f32(false, A10, false, B0, (short)0, acc1, false, false);
        acc1 = __builtin_amdgcn_wmma_f32_16x16x4_f32(false, A11, false, B1, (short)0, acc1, false, false);

        const float sc = scale[ech];
        const float sh = shift[ech];
        float mx = -3.402823466e38f;
#pragma unroll
        for (int j = 0; j < 8; ++j) mx = fmaxf(mx, hsw(acc0[j] * sc + sh));
        if (h == 0) {                       // acc1 rows 0..3 == k 16..19
#pragma unroll
            for (int j = 0; j < 4; ++j) mx = fmaxf(mx, hsw(acc1[j] * sc + sh));
        }
        mx = fmaxf(mx, __shfl_xor(mx, 16, 32));
        if (lane < 16) outx[((size_t)b * E_ + ech) * N_ + n] = mx;
    }
}

extern "C" void kernel_launch(void* const* d_in, const int* in_sizes, int n_in,
                              void* d_out, int out_size, void* d_ws, size_t ws_size,
                              hipStream_t stream) {
    const float* xyz   = (const float*)d_in[0];
    const float* W     = (const float*)d_in[1];
    const float* gamma = (const float*)d_in[2];
    const float* beta  = (const float*)d_in[3];
    float* out = (float*)d_out;

    // workspace layout: [idx: B*N*K ints | stats: 27 f32 | scale: 64 f32 | shift: 64 f32]
    int*   idxws = (int*)d_ws;
    float* stats = (float*)((char*)d_ws + (size_t)B_ * N_ * K_ * sizeof(int));
    float* scale = stats + 32;
    float* shift = scale + 64;

    hipMemsetAsync(stats, 0, 27 * sizeof(float), stream);
    knn_kernel     <<<dim3(N_ / 256, B_), 256, 0, stream>>>(xyz, idxws);
    stats_kernel   <<<1024, 256, 0, stream>>>(xyz, idxws, stats);
    finalize_kernel<<<1, E_, 0, stream>>>(W, gamma, beta, stats, scale, shift);
    edgeconv_kernel<<<B_ * N_, 32, 0, stream>>>(xyz, W, idxws, scale, shift, out);
}